// AttentionGRU_50783693308403
// MI455X (gfx1250) — compile-verified
//
#include <hip/hip_runtime.h>
#include <math.h>

// Problem constants (from reference)
#define NNODES   16384
#define NPAR     8192
#define NLEAF    8192
#define KWORDS   32
#define MAXDEG   4
#define VOCABSZ  100000
#define HID      64
#define NCLS     4
#define NEGINF   (-1e30f)

typedef __attribute__((ext_vector_type(2))) float v2f;
typedef __attribute__((ext_vector_type(8))) float v8f;

__device__ __forceinline__ float sigmoidf_(float x) { return 1.0f / (1.0f + __expf(-x)); }

// ---------------------------------------------------------------------------
// K0: transpose E (HID x VOCAB) -> Et (VOCAB x HID) so per-token gathers are
// contiguous 256B rows. Classic LDS tile transpose, both sides coalesced.
// ---------------------------------------------------------------------------
__global__ void k_transpose_embed(const float* __restrict__ E, float* __restrict__ Et) {
    __shared__ float tile[32][33];
    int v0 = blockIdx.x * 32;
    int h0 = blockIdx.y * 32;
    int tx = threadIdx.x, ty = threadIdx.y;  // 32 x 8
#pragma unroll
    for (int i = 0; i < 32; i += 8)
        tile[ty + i][tx] = E[(size_t)(h0 + ty + i) * VOCABSZ + (v0 + tx)];
    __syncthreads();
#pragma unroll
    for (int i = 0; i < 32; i += 8)
        Et[(size_t)(v0 + ty + i) * HID + (h0 + tx)] = tile[tx][ty + i];
}

// ---------------------------------------------------------------------------
// K1: xe[n,h] = sum_k x_word[n,k] * Et[x_index[n,k], h].
// One 64-thread block per node; lane h reads Et row element h (coalesced).
// Prefetch next row (global_prefetch_b8) while FMA-ing current one.
// ---------------------------------------------------------------------------
__global__ void k_embed_combine(const float* __restrict__ xw, const int* __restrict__ xi,
                                const float* __restrict__ Et, float* __restrict__ xe) {
    int n = blockIdx.x;
    int h = threadIdx.x;  // 0..63
    float acc = 0.f;
    int idx = xi[n * KWORDS];
    for (int k = 0; k < KWORDS; ++k) {
        float w = xw[n * KWORDS + k];
        int nidx = (k + 1 < KWORDS) ? xi[n * KWORDS + k + 1] : idx;
        __builtin_prefetch(&Et[(size_t)nidx * HID + h], 0, 0);
        acc = fmaf(w, Et[(size_t)idx * HID + h], acc);
        idx = nidx;
    }
    xe[n * HID + h] = acc;
}

// ---------------------------------------------------------------------------
// K2: dual 64x64 GEMM with V_WMMA_F32_16X16X4_F32 (exact fp32 path).
//   out{0,1}[n,h] = X[n,:] @ B{0,1} (+ bias), B = W^T if trans else W.
// Block = 256 thr (8 waves) handles a 16-row tile of X. Wave w computes
// matrix (w>>2), column tile (w&3)*16: 16 k-steps of WMMA (K=64 total).
// LDS: B padded to row stride 65, A tile stride 66 (bank-conflict free frags).
// ---------------------------------------------------------------------------
__global__ void __launch_bounds__(256) k_gemm_dual(
    const float* __restrict__ X,
    const float* __restrict__ W0, const float* __restrict__ W1,
    const float* __restrict__ bias0, const float* __restrict__ bias1,
    int trans0, int trans1,
    float* __restrict__ out0, float* __restrict__ out1) {
    __shared__ float sW[2][64 * 65];
    __shared__ float sX[16 * 66];
    __shared__ float sB[2][64];
    int tid = threadIdx.x;
    int r0 = blockIdx.x * 16;

    for (int e = tid; e < 4096; e += 256) {
        int r = e >> 6, c = e & 63;  // W[r][c]
        sW[0][trans0 ? (c * 65 + r) : (r * 65 + c)] = W0[e];  // B[k][n] layout
        sW[1][trans1 ? (c * 65 + r) : (r * 65 + c)] = W1[e];
    }
    for (int e = tid; e < 16 * HID; e += 256) {
        int r = e >> 6, c = e & 63;
        sX[r * 66 + c] = X[(r0 + r) * HID + c];
    }
    if (tid < 64) {
        sB[0][tid] = bias0 ? bias0[tid] : 0.f;
        sB[1][tid] = bias1 ? bias1[tid] : 0.f;
    }
    __syncthreads();

    int wv   = tid >> 5;          // wave 0..7
    int lane = tid & 31;
    int mi   = wv >> 2;           // which matrix
    int n0   = (wv & 3) * 16;     // output column tile
    int mrow = lane & 15;         // M (for A) / N (for B) index
    int koff = (lane >> 4) * 2;   // lanes 16-31 hold K+2,K+3

    const float* sw = sW[mi];
    v8f acc = {0.f, 0.f, 0.f, 0.f, 0.f, 0.f, 0.f, 0.f};
#pragma unroll
    for (int kk = 0; kk < 16; ++kk) {
        int kb = kk * 4 + koff;
        v2f a;
        a.x = sX[mrow * 66 + kb];
        a.y = sX[mrow * 66 + kb + 1];
        v2f b;
        b.x = sw[kb * 65 + (n0 + mrow)];
        b.y = sw[(kb + 1) * 65 + (n0 + mrow)];
        acc = __builtin_amdgcn_wmma_f32_16x16x4_f32(false, a, false, b, (short)0, acc,
                                                    false, false);
    }
    float* outp = mi ? out1 : out0;
    int colg  = n0 + mrow;
    float bv  = sB[mi][colg];
    int mbase = (lane >> 4) * 8;
#pragma unroll
    for (int v = 0; v < 8; ++v) {
        int row = r0 + mbase + v;
        outp[row * HID + colg] = acc[v] + bv;
    }
}

// ---------------------------------------------------------------------------
// K2b: leaf GRU init: node_h = (1 - sigmoid(GZ)) * tanh(GH), rows [0, L).
// ---------------------------------------------------------------------------
__global__ void k_leaf(const float* __restrict__ GZ, const float* __restrict__ GH,
                       float* __restrict__ node_h) {
    int i = blockIdx.x * 256 + threadIdx.x;  // < NLEAF*HID
    float z = sigmoidf_(GZ[i]);
    float c = tanhf(GH[i]);
    node_h[i] = (1.f - z) * c;
}

// ---------------------------------------------------------------------------
// K3: the sequential scan — single wave32 carries the whole 8192-step chain.
// Per step only the node_h-dependent work remains: gather <=4 child vectors,
// 4 attention dots (shfl_xor reductions), softmax, h_tilde, then 3 64x64
// U-matvecs via LDS-resident U (row stride 65 -> conflict-free per-lane rows).
// Single-wave workgroup: barriers are S_NOPs; DS ops are in-order in-wave.
// ---------------------------------------------------------------------------
__global__ void __launch_bounds__(32) k_scan(
    const int* __restrict__ tree,
    const float* __restrict__ GZ, const float* __restrict__ GR,
    const float* __restrict__ GH, const float* __restrict__ GA,
    const float* __restrict__ Uz, const float* __restrict__ Ur,
    const float* __restrict__ Uh,
    float* __restrict__ node_h) {
    __shared__ float sUz[64 * 65], sUr[64 * 65], sUh[64 * 65];
    __shared__ float sHT[64], sRH[64];
    int lane = threadIdx.x;
    for (int e = lane; e < 4096; e += 32) {
        int r = e >> 6, c = e & 63;
        sUz[r * 65 + c] = Uz[e];
        sUr[r * 65 + c] = Ur[e];
        sUh[r * 65 + c] = Uh[e];
    }
    __syncthreads();
    int h0 = lane, h1 = lane + 32;

    for (int p = 0; p < NPAR; ++p) {
        int row = NLEAF + p;
        int c0 = tree[p * (MAXDEG + 1) + 0];
        int c1 = tree[p * (MAXDEG + 1) + 1];
        int c2 = tree[p * (MAXDEG + 1) + 2];
        int c3 = tree[p * (MAXDEG + 1) + 3];
        int i0 = c0 > -1 ? c0 : 0, i1 = c1 > -1 ? c1 : 0;
        int i2 = c2 > -1 ? c2 : 0, i3 = c3 > -1 ? c3 : 0;

        float a0 = GA[row * HID + h0], a1 = GA[row * HID + h1];
        float pz0 = GZ[row * HID + h0], pz1 = GZ[row * HID + h1];
        float pr0 = GR[row * HID + h0], pr1 = GR[row * HID + h1];
        float ph0 = GH[row * HID + h0], ph1 = GH[row * HID + h1];

        float x00 = node_h[i0 * HID + h0], x01 = node_h[i0 * HID + h1];
        float x10 = node_h[i1 * HID + h0], x11 = node_h[i1 * HID + h1];
        float x20 = node_h[i2 * HID + h0], x21 = node_h[i2 * HID + h1];
        float x30 = node_h[i3 * HID + h0], x31 = node_h[i3 * HID + h1];

        float d0 = a0 * x00 + a1 * x01;
        float d1 = a0 * x10 + a1 * x11;
        float d2 = a0 * x20 + a1 * x21;
        float d3 = a0 * x30 + a1 * x31;
#pragma unroll
        for (int off = 16; off > 0; off >>= 1) {
            d0 += __shfl_xor(d0, off, 32);
            d1 += __shfl_xor(d1, off, 32);
            d2 += __shfl_xor(d2, off, 32);
            d3 += __shfl_xor(d3, off, 32);
        }
        float l0 = (c0 > -1) ? sigmoidf_(d0) : NEGINF;
        float l1 = (c1 > -1) ? sigmoidf_(d1) : NEGINF;
        float l2 = (c2 > -1) ? sigmoidf_(d2) : NEGINF;
        float l3 = (c3 > -1) ? sigmoidf_(d3) : NEGINF;
        float mx = fmaxf(fmaxf(l0, l1), fmaxf(l2, l3));
        float e0 = __expf(l0 - mx), e1 = __expf(l1 - mx);
        float e2 = __expf(l2 - mx), e3 = __expf(l3 - mx);
        float inv = 1.f / (e0 + e1 + e2 + e3);
        float t0 = e0 * inv, t1 = e1 * inv, t2 = e2 * inv, t3 = e3 * inv;

        float ht0 = t0 * x00 + t1 * x10 + t2 * x20 + t3 * x30;
        float ht1 = t0 * x01 + t1 * x11 + t2 * x21 + t3 * x31;
        sHT[h0] = ht0;
        sHT[h1] = ht1;
        __syncthreads();

        float az0 = 0.f, az1 = 0.f, ar0 = 0.f, ar1 = 0.f;
#pragma unroll 8
        for (int k = 0; k < 64; ++k) {
            float htk = sHT[k];
            az0 = fmaf(sUz[h0 * 65 + k], htk, az0);
            az1 = fmaf(sUz[h1 * 65 + k], htk, az1);
            ar0 = fmaf(sUr[h0 * 65 + k], htk, ar0);
            ar1 = fmaf(sUr[h1 * 65 + k], htk, ar1);
        }
        float z0 = sigmoidf_(pz0 + az0), z1 = sigmoidf_(pz1 + az1);
        float r0 = sigmoidf_(pr0 + ar0), r1 = sigmoidf_(pr1 + ar1);
        sRH[h0] = r0 * ht0;
        sRH[h1] = r1 * ht1;
        __syncthreads();

        float ac0 = 0.f, ac1 = 0.f;
#pragma unroll 8
        for (int k = 0; k < 64; ++k) {
            float rhk = sRH[k];
            ac0 = fmaf(sUh[h0 * 65 + k], rhk, ac0);
            ac1 = fmaf(sUh[h1 * 65 + k], rhk, ac1);
        }
        float cc0 = tanhf(ph0 + ac0), cc1 = tanhf(ph1 + ac1);
        float hh0 = z0 * ht0 + (1.f - z0) * cc0;
        float hh1 = z1 * ht1 + (1.f - z1) * cc1;
        node_h[row * HID + h0] = hh0;
        node_h[row * HID + h1] = hh1;
        __syncthreads();  // guards sHT/sRH reuse; S_NOP for single-wave WG
    }
}

// ---------------------------------------------------------------------------
// K4: final = max over parent rows; out = softmax(W_out*final row-sums + b).
// ---------------------------------------------------------------------------
__global__ void __launch_bounds__(256) k_output(const float* __restrict__ node_h,
                                                const float* __restrict__ Wout,
                                                const float* __restrict__ bout,
                                                float* __restrict__ out) {
    __shared__ float pm[256];
    __shared__ float fm[64];
    __shared__ float lg[NCLS];
    int tid = threadIdx.x;
    int col = tid & 63, seg = tid >> 6;
    float m = -INFINITY;
    for (int p = seg; p < NPAR; p += 4)
        m = fmaxf(m, node_h[(NLEAF + p) * HID + col]);
    pm[tid] = m;
    __syncthreads();
    if (tid < 64)
        fm[tid] = fmaxf(fmaxf(pm[tid], pm[tid + 64]), fmaxf(pm[tid + 128], pm[tid + 192]));
    __syncthreads();
    if (tid < NCLS) {
        float s = bout[tid];
        for (int h = 0; h < HID; ++h) s = fmaf(Wout[tid * HID + h], fm[h], s);
        lg[tid] = s;
    }
    __syncthreads();
    if (tid == 0) {
        float mx = fmaxf(fmaxf(lg[0], lg[1]), fmaxf(lg[2], lg[3]));
        float e0 = __expf(lg[0] - mx), e1 = __expf(lg[1] - mx);
        float e2 = __expf(lg[2] - mx), e3 = __expf(lg[3] - mx);
        float inv = 1.f / (e0 + e1 + e2 + e3);
        out[0] = e0 * inv; out[1] = e1 * inv; out[2] = e2 * inv; out[3] = e3 * inv;
    }
}

// ---------------------------------------------------------------------------
extern "C" void kernel_launch(void* const* d_in, const int* in_sizes, int n_in,
                              void* d_out, int out_size, void* d_ws, size_t ws_size,
                              hipStream_t stream) {
    (void)in_sizes; (void)n_in; (void)out_size; (void)ws_size;
    const float* xw   = (const float*)d_in[0];
    const int*   xi   = (const int*)d_in[1];
    const int*   tree = (const int*)d_in[2];
    const float* E    = (const float*)d_in[3];
    const float* Wz   = (const float*)d_in[4];
    const float* Uz   = (const float*)d_in[5];
    const float* bz   = (const float*)d_in[6];
    const float* Wr   = (const float*)d_in[7];
    const float* Ur   = (const float*)d_in[8];
    const float* br   = (const float*)d_in[9];
    const float* Wh   = (const float*)d_in[10];
    const float* Uh   = (const float*)d_in[11];
    const float* bh   = (const float*)d_in[12];
    const float* Wa   = (const float*)d_in[13];
    const float* Wout = (const float*)d_in[14];
    const float* bout = (const float*)d_in[15];
    float* out = (float*)d_out;

    // Workspace layout (floats): ~50.8 MB total
    float* p      = (float*)d_ws;
    float* Et     = p; p += (size_t)VOCABSZ * HID;  // 25.6 MB
    float* xe     = p; p += (size_t)NNODES * HID;   // 4 MB
    float* GZ     = p; p += (size_t)NNODES * HID;
    float* GR     = p; p += (size_t)NNODES * HID;
    float* GH     = p; p += (size_t)NNODES * HID;
    float* GA     = p; p += (size_t)NNODES * HID;
    float* node_h = p; p += (size_t)NNODES * HID;

    k_transpose_embed<<<dim3(VOCABSZ / 32, HID / 32), dim3(32, 8), 0, stream>>>(E, Et);
    k_embed_combine<<<NNODES, HID, 0, stream>>>(xw, xi, Et, xe);
    k_gemm_dual<<<NNODES / 16, 256, 0, stream>>>(xe, Wz, Wr, bz, br, 1, 1, GZ, GR);
    k_gemm_dual<<<NNODES / 16, 256, 0, stream>>>(xe, Wh, Wa, bh, nullptr, 1, 0, GH, GA);
    k_leaf<<<(NLEAF * HID) / 256, 256, 0, stream>>>(GZ, GH, node_h);
    k_scan<<<1, 32, 0, stream>>>(tree, GZ, GR, GH, GA, Uz, Ur, Uh, node_h);
    k_output<<<1, 256, 0, stream>>>(node_h, Wout, bout, out);
}